// GNN_16913581212178
// MI455X (gfx1250) — compile-verified
//
#include <hip/hip_runtime.h>

typedef __attribute__((ext_vector_type(2))) float v2f;
typedef __attribute__((ext_vector_type(8))) float v8f;

// Dims fixed by the reference
#define IN_DIM  64
#define HIDDEN  64
#define OUT_DIM 2

// ---------------------------------------------------------------------------
// Kernel 1: zero the workspace (msg_sum [N*64] and deg [N])
// ---------------------------------------------------------------------------
__global__ void gnn_zero_ws(float* __restrict__ p, long n) {
    long i = (long)blockIdx.x * blockDim.x + threadIdx.x;
    long stride = (long)gridDim.x * blockDim.x;
    for (; i < n; i += stride) p[i] = 0.0f;
}

// ---------------------------------------------------------------------------
// Kernel 2: edge scatter-add. 64 threads (2 waves) per edge; lane k handles
// feature k. Reads of x[src] and atomic adds to msg[dst] are 256B-contiguous
// per edge, so this is L2-resident streaming with f32 atomics.
// ---------------------------------------------------------------------------
__global__ __launch_bounds__(256) void gnn_edge_agg(
    const float* __restrict__ x,
    const int*   __restrict__ src,
    const int*   __restrict__ dst,
    float* __restrict__ msg,
    float* __restrict__ deg,
    long E) {
    long t = (long)blockIdx.x * blockDim.x + threadIdx.x;
    if (t >= E * 64) return;
    long e = t >> 6;
    int  k = (int)(t & 63);
    int  s = src[e];
    int  d = dst[e];
    atomicAdd(&msg[(long)d * IN_DIM + k], x[(long)s * IN_DIM + k]);
    if (k == 0) atomicAdd(&deg[d], 1.0f);
}

// ---------------------------------------------------------------------------
// Kernel 3: per-16-node-tile SAGE update with fp32 WMMA.
//   h = (msg/deg) @ Wl^T + x @ Wr^T + bl ; z = relu(h) ; out = z @ Wh^T + bh
// One wave (32 lanes) owns one 16-row tile. A-fragment layout (16x4 f32):
//   lane L -> row L%16 ; VGPR0/1 hold K = 4*kb + (L<16 ? 0,1 : 2,3)
// B-fragment (4x16): lane L -> col L%16 ; same K split across lane halves.
// D/C (16x16 f32, v8f): element v -> row v + (L<16?0:8), col L%16.
// EXEC stays all-ones around WMMA; out-of-range rows use zero scaling.
// ---------------------------------------------------------------------------
__global__ __launch_bounds__(256) void gnn_node_update(
    const float* __restrict__ x,
    const float* __restrict__ msg,
    const float* __restrict__ deg,
    const float* __restrict__ Wl,  // [64,64] row-major (HIDDEN, IN)
    const float* __restrict__ bl,  // [64]
    const float* __restrict__ Wr,  // [64,64]
    const float* __restrict__ Wh,  // [2,64]
    const float* __restrict__ bh,  // [2]
    float* __restrict__ out,       // [N,2]
    int N) {
    __shared__ float zbuf[8][16 * HIDDEN];   // 8 waves * 16x64 z tile = 32 KB

    const int lane = threadIdx.x & 31;
    const int wave = threadIdx.x >> 5;
    const int tile = blockIdx.x * 8 + wave;
    const int r    = lane & 15;       // row-in-tile (A) / column (B,C,D)
    const int hi   = lane >> 4;       // 0: lanes 0-15, 1: lanes 16-31

    const int  row   = tile * 16 + r;
    const bool valid = (row < N);
    const int  srow  = valid ? row : 0;

    const float2* xrow = (const float2*)(x   + (long)srow * IN_DIM);
    const float2* mrow = (const float2*)(msg + (long)srow * IN_DIM);
    const float scale = valid ? (1.0f / fmaxf(deg[srow], 1.0f)) : 0.0f;
    const float xs    = valid ? 1.0f : 0.0f;

    // element-pair offset inside a 4-wide K step: lanes 0-15 -> pair 0, 16-31 -> pair 1
    const int koff2 = hi;

    for (int j = 0; j < 4; ++j) {           // 16-wide hidden blocks
        const int n = j * 16 + r;           // B/D column = hidden index
        const float2* wlrow = (const float2*)(Wl + (long)n * IN_DIM);
        const float2* wrrow = (const float2*)(Wr + (long)n * IN_DIM);

        v8f c = {0.f, 0.f, 0.f, 0.f, 0.f, 0.f, 0.f, 0.f};

        #pragma unroll
        for (int kb = 0; kb < 16; ++kb) {   // K = 64 in steps of 4
            const int p = 2 * kb + koff2;
            float2 am = mrow[p];
            float2 ax = xrow[p];
            float2 wl = wlrow[p];
            float2 wr = wrrow[p];

            v2f aA; aA.x = am.x * scale; aA.y = am.y * scale;
            v2f aX; aX.x = ax.x * xs;    aX.y = ax.y * xs;
            v2f bL; bL.x = wl.x;         bL.y = wl.y;
            v2f bR; bR.x = wr.x;         bR.y = wr.y;

            c = __builtin_amdgcn_wmma_f32_16x16x4_f32(
                    false, aA, false, bL, (short)0, c, false, false);
            c = __builtin_amdgcn_wmma_f32_16x16x4_f32(
                    false, aX, false, bR, (short)0, c, false, false);
        }

        const float bias = bl[n];
        #pragma unroll
        for (int v = 0; v < 8; ++v) {
            float h = c[v] + bias;          // row v + 8*hi, col n
            h = fmaxf(h, 0.0f);             // ReLU
            zbuf[wave][(v + hi * 8) * HIDDEN + n] = h;
        }
    }

    // Head: OUT_DIM = 2. Lanes 0-15 -> output 0, lanes 16-31 -> output 1,
    // each lane reduces one row's 64 hidden values from LDS.
    float acc = bh[hi];
    #pragma unroll 8
    for (int nn = 0; nn < HIDDEN; ++nn)
        acc += zbuf[wave][r * HIDDEN + nn] * Wh[hi * HIDDEN + nn];

    if (valid) out[(long)row * OUT_DIM + hi] = acc;
}

// ---------------------------------------------------------------------------
// Launch
// ---------------------------------------------------------------------------
extern "C" void kernel_launch(void* const* d_in, const int* in_sizes, int n_in,
                              void* d_out, int out_size, void* d_ws, size_t ws_size,
                              hipStream_t stream) {
    const float* x  = (const float*)d_in[0];
    const int*   ei = (const int*)d_in[1];     // [2,E] flat, int32
    const float* Wl = (const float*)d_in[2];
    const float* bl = (const float*)d_in[3];
    const float* Wr = (const float*)d_in[4];
    const float* Wh = (const float*)d_in[5];
    const float* bh = (const float*)d_in[6];
    float* out = (float*)d_out;

    const long N = in_sizes[0] / IN_DIM;
    const long E = in_sizes[1] / 2;
    const int* src = ei;
    const int* dst = ei + E;

    float* msg = (float*)d_ws;            // [N * 64]
    float* deg = msg + (size_t)N * IN_DIM; // [N]

    // 1) zero msg + deg
    {
        long n = N * IN_DIM + N;
        int blocks = (int)((n + 256 * 8 - 1) / (256 * 8));
        if (blocks < 1) blocks = 1;
        gnn_zero_ws<<<blocks, 256, 0, stream>>>(msg, n);
    }
    // 2) edge scatter-add
    {
        long threads = E * 64;
        int blocks = (int)((threads + 255) / 256);
        gnn_edge_agg<<<blocks, 256, 0, stream>>>(x, src, dst, msg, deg, E);
    }
    // 3) WMMA node update + head
    {
        long tiles  = (N + 15) / 16;
        int  blocks = (int)((tiles + 7) / 8);
        gnn_node_update<<<blocks, 256, 0, stream>>>(
            x, msg, deg, Wl, bl, Wr, Wh, bh, out, (int)N);
    }
}